// GCAB_27900107555233
// MI455X (gfx1250) — compile-verified
//
#include <hip/hip_runtime.h>
#include <hip/hip_bf16.h>

typedef __attribute__((ext_vector_type(2))) float v2f;
typedef __attribute__((ext_vector_type(8))) float v8f;

#define NNODE 52
#define NEDGE 832
#define NE2   884   // 832 + 52 self loops
#define FDIM  256
#define NGRAPH 512
#define MROWS (NGRAPH * NNODE)   // 26624, divisible by 128

__device__ __forceinline__ v8f wmma_f32_16x16x4(v2f a, v2f b, v8f c) {
    // 8 args: (neg_a, A, neg_b, B, c_mod, C, reuse_a, reuse_b)
    return __builtin_amdgcn_wmma_f32_16x16x4_f32(
        false, a, false, b, (short)0, c, false, false);
}

__device__ __forceinline__ void atomicMaxFloatLds(float* addr, float value) {
    // classic signed/unsigned bit-trick; valid for mixed signs with -inf init
    if (value >= 0.0f)
        atomicMax((int*)addr, __float_as_int(value));
    else
        atomicMin((unsigned int*)addr, __float_as_uint(value));
}

// ---------------------------------------------------------------------------
// Y[M,256] = X[M,256] @ W[256,256].
// Each wave computes a 16x128 strip (8 N-tiles) so each A fragment is reused
// 8x -> X L2 traffic cut 8x. Block = 8 waves = 128 rows; grid=(M/128, 256/128).
// ---------------------------------------------------------------------------
__global__ __launch_bounds__(256) void gemm_wmma(
    const float* __restrict__ X, const float* __restrict__ W,
    float* __restrict__ Y) {
    const int lane = threadIdx.x & 31;
    const int wave = threadIdx.x >> 5;
    const int mBase = (blockIdx.x * 8 + wave) * 16;
    const int nBase = blockIdx.y * 128;          // 8 tiles of 16 columns
    const int r  = lane & 15;
    const int hi = lane >> 4;
    const int kh = hi * 2;

    v8f c[8];
#pragma unroll
    for (int t = 0; t < 8; ++t) c[t] = (v8f){};

    const float* aPtr = X + (size_t)(mBase + r) * FDIM + kh;
    const float* bPtr = W + (size_t)kh * FDIM + nBase + r;

    for (int kg = 0; kg < FDIM; kg += 16) {      // 16-deep K groups
        // prefetch next K group of A row and W panel (global_prefetch_b8)
        __builtin_prefetch(aPtr + 16, 0, 0);
        __builtin_prefetch(bPtr + 16 * FDIM, 0, 0);
#pragma unroll
        for (int kk = 0; kk < 16; kk += 4) {
            v2f a = *(const v2f*)(aPtr + kk);    // A: M=lane&15, K pair kh
            const float* bp = bPtr + (size_t)kk * FDIM;
#pragma unroll
            for (int t = 0; t < 8; ++t) {
                v2f b;
                b.x = bp[t * 16];                // B row k+kh   , col tile t
                b.y = bp[t * 16 + FDIM];         // B row k+kh+1
                c[t] = wmma_f32_16x16x4(a, b, c[t]);
            }
        }
        aPtr += 16;
        bPtr += (size_t)16 * FDIM;
    }

#pragma unroll
    for (int t = 0; t < 8; ++t) {
#pragma unroll
        for (int v = 0; v < 8; ++v) {
            const int row = mBase + v + 8 * hi;  // C layout: vgpr v, half hi
            Y[(size_t)row * FDIM + nBase + t * 16 + r] = c[t][v];
        }
    }
}

// ---------------------------------------------------------------------------
// Per-graph GAT attention + WMMA aggregation.
//   OUT = softmax-aggregate(XP) + bias  [ + res, relu ]  (relu_mode)
// grid = 512 blocks (one per graph), block = 256.
// ---------------------------------------------------------------------------
__global__ __launch_bounds__(256) void gat_attn(
    const float* __restrict__ XP, const int* __restrict__ EI,
    const float* __restrict__ a_s, const float* __restrict__ a_d,
    const float* __restrict__ bias, const float* __restrict__ res,
    float* __restrict__ OUT, int relu_mode) {
    __shared__ float Attn[64 * NNODE];     // padded rows 52..63 stay zero
    __shared__ float ebuf[NE2];
    __shared__ float asrc[NNODE], adst[NNODE], emax[NNODE], denom[NNODE];

    const int g    = blockIdx.x;
    const int tid  = threadIdx.x;
    const int lane = tid & 31;
    const int wave = tid >> 5;
    const float* xp  = XP + (size_t)g * NNODE * FDIM;
    const int*   src = EI + (size_t)g * 2 * NEDGE;
    const int*   dst = src + NEDGE;

    for (int i = tid; i < 64 * NNODE; i += 256) Attn[i] = 0.0f;
    if (tid < NNODE) { emax[tid] = -__builtin_huge_valf(); denom[tid] = 0.0f; }

    // asrc/adst: one node per wave, shuffle-reduce over 32 lanes (wave32)
    for (int n = wave; n < NNODE; n += 8) {
        float ss = 0.0f, dd = 0.0f;
        for (int f = lane; f < FDIM; f += 32) {
            float v = xp[n * FDIM + f];
            ss += v * a_s[f];
            dd += v * a_d[f];
        }
        for (int off = 16; off > 0; off >>= 1) {
            ss += __shfl_xor(ss, off, 32);
            dd += __shfl_xor(dd, off, 32);
        }
        if (lane == 0) { asrc[n] = ss; adst[n] = dd; }
    }
    __syncthreads();

    // pass 1: leaky-relu logits + segment max over dst
    for (int e = tid; e < NE2; e += 256) {
        int s = (e < NEDGE) ? src[e] : (e - NEDGE);
        int d = (e < NEDGE) ? dst[e] : (e - NEDGE);
        float v = asrc[s] + adst[d];
        v = (v > 0.0f) ? v : 0.2f * v;
        ebuf[e] = v;
        atomicMaxFloatLds(&emax[d], v);
    }
    __syncthreads();
    // pass 2: exp + segment sum
    for (int e = tid; e < NE2; e += 256) {
        int d = (e < NEDGE) ? dst[e] : (e - NEDGE);
        float w = __expf(ebuf[e] - emax[d]);
        ebuf[e] = w;
        atomicAdd(&denom[d], w);
    }
    __syncthreads();
    // pass 3: scatter alpha into dense [dst, src] matrix
    for (int e = tid; e < NE2; e += 256) {
        int s = (e < NEDGE) ? src[e] : (e - NEDGE);
        int d = (e < NEDGE) ? dst[e] : (e - NEDGE);
        atomicAdd(&Attn[d * NNODE + s], ebuf[e] / denom[d]);
    }
    __syncthreads();

    // OUT[52,256] = Attn[52x52(pad 64)] @ xp[52,256]; 4 M-tiles x 16 N-tiles
    const int r  = lane & 15;
    const int hi = lane >> 4;
    const int kh = hi * 2;
    for (int t = wave; t < 64; t += 8) {
        const int mBase = (t & 3) * 16;
        const int nBase = (t >> 2) * 16;
        v8f c = {};
#pragma unroll
        for (int k0 = 0; k0 < NNODE; k0 += 4) {   // 52 = 13 * 4 exactly
            v2f a, b;
            a.x = Attn[(mBase + r) * NNODE + k0 + kh];
            a.y = Attn[(mBase + r) * NNODE + k0 + kh + 1];
            b.x = xp[(k0 + kh) * FDIM + nBase + r];
            b.y = xp[(k0 + kh + 1) * FDIM + nBase + r];
            c = wmma_f32_16x16x4(a, b, c);
        }
#pragma unroll
        for (int v = 0; v < 8; ++v) {
            const int row = mBase + v + 8 * hi;
            if (row < NNODE) {
                const int col = nBase + r;
                size_t idx = (size_t)g * NNODE * FDIM + (size_t)row * FDIM + col;
                float o = c[v] + bias[col];
                if (relu_mode) {
                    o += res[idx];
                    o = (o > 0.0f) ? o : 0.0f;
                }
                OUT[idx] = o;
            }
        }
    }
}

// ---------------------------------------------------------------------------
// Per-graph avg/max pooling over the node axis -> POOL rows [2g]=avg, [2g+1]=max
// ---------------------------------------------------------------------------
__global__ __launch_bounds__(256) void pool_kernel(
    const float* __restrict__ H, float* __restrict__ POOL) {
    const int g = blockIdx.x;
    const int f = threadIdx.x;
    const float* h = H + (size_t)g * NNODE * FDIM;
    float s = 0.0f, m = -__builtin_huge_valf();
    for (int n = 0; n < NNODE; ++n) {
        float v = h[n * FDIM + f];
        s += v;
        m = fmaxf(m, v);
    }
    POOL[(size_t)(2 * g) * FDIM + f]     = s * (1.0f / (float)NNODE);
    POOL[(size_t)(2 * g + 1) * FDIM + f] = m;
}

// ---------------------------------------------------------------------------
// Channel gate + per-node max + 1->1 GCN + sigmoid node gate + final product.
// ---------------------------------------------------------------------------
__global__ __launch_bounds__(256) void final_kernel(
    const float* __restrict__ H, const int* __restrict__ EI,
    const float* __restrict__ PM, const float* __restrict__ bm,
    const float* __restrict__ wg, const float* __restrict__ bg,
    float* __restrict__ OUT) {
    __shared__ float mch[FDIM];
    __shared__ float hmax[NNODE], deg[NNODE], agg[NNODE], mno[NNODE];

    const int g    = blockIdx.x;
    const int tid  = threadIdx.x;
    const int lane = tid & 31;
    const int wave = tid >> 5;
    const float* h   = H + (size_t)g * NNODE * FDIM;
    const int*   src = EI + (size_t)g * 2 * NEDGE;
    const int*   dst = src + NEDGE;

    {   // channel attention gate
        const int f = tid;
        float p0 = PM[(size_t)(2 * g) * FDIM + f] + bm[f];
        float p1 = PM[(size_t)(2 * g + 1) * FDIM + f] + bm[f];
        p0 = fmaxf(p0, 0.0f);
        p1 = fmaxf(p1, 0.0f);
        mch[f] = 1.0f / (1.0f + __expf(-(p0 + p1)));
    }
    if (tid < NNODE) { deg[tid] = 1.0f; agg[tid] = 0.0f; }   // self-loop deg
    __syncthreads();

    // per-node max over gated features
    for (int n = wave; n < NNODE; n += 8) {
        float m = -__builtin_huge_valf();
        for (int f = lane; f < FDIM; f += 32)
            m = fmaxf(m, h[n * FDIM + f] * mch[f]);
        for (int off = 16; off > 0; off >>= 1)
            m = fmaxf(m, __shfl_xor(m, off, 32));
        if (lane == 0) hmax[n] = m;
    }
    // in-degree (dst occurrences)
    for (int e = tid; e < NEDGE; e += 256) atomicAdd(&deg[dst[e]], 1.0f);
    __syncthreads();

    const float w = wg[0], b = bg[0];
    for (int e = tid; e < NE2; e += 256) {
        int s = (e < NEDGE) ? src[e] : (e - NEDGE);
        int d = (e < NEDGE) ? dst[e] : (e - NEDGE);
        float norm = rsqrtf(deg[s]) * rsqrtf(deg[d]);
        atomicAdd(&agg[d], norm * hmax[s] * w);
    }
    __syncthreads();
    if (tid < NNODE) mno[tid] = 1.0f / (1.0f + __expf(-(agg[tid] + b)));
    __syncthreads();

    for (int i = tid; i < NNODE * FDIM; i += 256) {
        int n = i >> 8;
        int f = i & 255;
        OUT[(size_t)g * NNODE * FDIM + i] = h[i] * mch[f] * mno[n];
    }
}

extern "C" void kernel_launch(void* const* d_in, const int* in_sizes, int n_in,
                              void* d_out, int out_size, void* d_ws, size_t ws_size,
                              hipStream_t stream) {
    const float* x    = (const float*)d_in[0];
    const int*   ei   = (const int*)d_in[1];
    const float* W0   = (const float*)d_in[2];
    const float* a_s0 = (const float*)d_in[3];
    const float* a_d0 = (const float*)d_in[4];
    const float* b0   = (const float*)d_in[5];
    const float* W1   = (const float*)d_in[6];
    const float* a_s1 = (const float*)d_in[7];
    const float* a_d1 = (const float*)d_in[8];
    const float* b1   = (const float*)d_in[9];
    const float* Wm   = (const float*)d_in[10];
    const float* bm   = (const float*)d_in[11];
    const float* wg   = (const float*)d_in[12];
    const float* bg   = (const float*)d_in[13];

    float* ws   = (float*)d_ws;
    float* XP   = ws;                          // [26624,256] (reused for both GATs)
    float* H0   = XP + (size_t)MROWS * FDIM;   // [26624,256]
    float* H1   = H0 + (size_t)MROWS * FDIM;   // [26624,256]
    float* POOL = H1 + (size_t)MROWS * FDIM;   // [1024,256]
    float* PM   = POOL + (size_t)1024 * FDIM;  // [1024,256]

    const dim3 gemmGrid(MROWS / 128, 2);       // 128-row x 128-col blocks
    const dim3 poolGemmGrid(1024 / 128, 2);

    // GAT0
    gemm_wmma<<<gemmGrid, 256, 0, stream>>>(x, W0, XP);
    gat_attn<<<NGRAPH, 256, 0, stream>>>(XP, ei, a_s0, a_d0, b0, nullptr, H0, 0);
    // GAT1 + residual + relu
    gemm_wmma<<<gemmGrid, 256, 0, stream>>>(H0, W1, XP);
    gat_attn<<<NGRAPH, 256, 0, stream>>>(XP, ei, a_s1, a_d1, b1, H0, H1, 1);
    // channel attention matvecs as a batched GEMM
    pool_kernel<<<NGRAPH, 256, 0, stream>>>(H1, POOL);
    gemm_wmma<<<poolGemmGrid, 256, 0, stream>>>(POOL, Wm, PM);
    // gates + GCN + output
    final_kernel<<<NGRAPH, 256, 0, stream>>>(H1, ei, PM, bm, wg, bg, (float*)d_out);
}